// SlidingWindowL2MaxPooling_6107443495474
// MI455X (gfx1250) — compile-verified
//
#include <hip/hip_runtime.h>

typedef __attribute__((ext_vector_type(2))) float v2f;
typedef __attribute__((ext_vector_type(8))) float v8f;

#define HAS_WMMA_F32_16X16X4 __has_builtin(__builtin_amdgcn_wmma_f32_16x16x4_f32)

namespace {
constexpr int S  = 2048;  // sequence length
constexpr int W  = 64;    // window / feature width
constexpr int NT = 256;   // threads per block (8 wave32s)
}

__global__ __launch_bounds__(NT) void swl2max_kernel(const float* __restrict__ X,
                                                     float* __restrict__ out) {
  __shared__ float sq[S];     // per-row sum of squares, later inclusive prefix
  __shared__ float part[NT];  // per-thread partials for scan
  __shared__ float bv[NT];    // argmax reduce: values
  __shared__ int   bi[NT];    // argmax reduce: indices

  const int t    = threadIdx.x;
  const int lane = t & 31;
  const int wave = t >> 5;
  const int m    = lane & 15;  // row within 16-row batch
  const int half = lane >> 4;  // 0: cols 0..31, 1: cols 32..63

  const size_t bf = blockIdx.x;               // 0 .. B*F-1 (2048)
  const float* Xb = X + bf * (size_t)(S * W); // 128K floats per (b,f)

  // ---------------- Phase 1: sq[s] = sum_w X[s,w]^2 ----------------
  // Each wave handles 16 rows per batch; lane m covers row m cols [0,32),
  // lane m+16 covers row m cols [32,64). VALU FMAs build per-lane partials,
  // then one V_WMMA_F32_16X16X4_F32 with B=ones reduces the two K-halves
  // per row in full f32 precision (D[m,*] = A[m,0]+A[m,2] = rowsum).
  for (int batch = wave; batch < S / 16; batch += NT / 32) {
    const int row0 = batch * 16;
    const float* p = Xb + (size_t)(row0 + m) * W + half * 32;
    float ps = 0.f;
#pragma unroll
    for (int kc = 0; kc < 8; ++kc) {
      float4 q = *reinterpret_cast<const float4*>(p + kc * 4);
      ps = fmaf(q.x, q.x, ps);
      ps = fmaf(q.y, q.y, ps);
      ps = fmaf(q.z, q.z, ps);
      ps = fmaf(q.w, q.w, ps);
    }
#if HAS_WMMA_F32_16X16X4
    v2f a; a.x = ps;  a.y = 0.f;   // A[m, K=0/2] = partial, A[m, K=1/3] = 0
    v2f b; b.x = 1.f; b.y = 1.f;   // B = ones (layout-invariant)
    v8f c = {0.f, 0.f, 0.f, 0.f, 0.f, 0.f, 0.f, 0.f};
    c = __builtin_amdgcn_wmma_f32_16x16x4_f32(
        /*neg_a=*/false, a, /*neg_b=*/false, b,
        /*c_mod=*/(short)0, c, /*reuse_a=*/false, /*reuse_b=*/false);
    // C/D layout: VGPR r holds M=r (lanes 0-15) and M=r+8 (lanes 16-31),
    // every column N identical -> lane 0 has rows 0..7, lane 16 rows 8..15.
    if (lane == 0) {
      *reinterpret_cast<float4*>(&sq[row0 + 0]) = make_float4(c[0], c[1], c[2], c[3]);
      *reinterpret_cast<float4*>(&sq[row0 + 4]) = make_float4(c[4], c[5], c[6], c[7]);
    } else if (lane == 16) {
      *reinterpret_cast<float4*>(&sq[row0 + 8])  = make_float4(c[0], c[1], c[2], c[3]);
      *reinterpret_cast<float4*>(&sq[row0 + 12]) = make_float4(c[4], c[5], c[6], c[7]);
    }
#else
    float tot = ps + __shfl_xor(ps, 16, 32);
    if (lane < 16) sq[row0 + m] = tot;
#endif
  }
  __syncthreads();

  // ---------------- Phase 2: prefix sum of sq -> inclusive cumsum ----------------
  float myP = 0.f;
#pragma unroll
  for (int j = 0; j < 8; ++j) myP += sq[t * 8 + j];
  part[t] = myP;
  __syncthreads();

  // Hillis-Steele inclusive scan of the 256 partials in LDS.
  for (int d = 1; d < NT; d <<= 1) {
    float addv = (t >= d) ? part[t - d] : 0.f;
    __syncthreads();
    part[t] += addv;
    __syncthreads();
  }

  // In-place: sq[i] -> inclusive prefix sum cs[i] (each thread owns its 8 slots).
  {
    float run = part[t] - myP;  // exclusive offset for this chunk
#pragma unroll
    for (int j = 0; j < 8; ++j) {
      run += sq[t * 8 + j];
      sq[t * 8 + j] = run;
    }
  }
  __syncthreads();

  // ---------------- Phase 3: windowed sums + first-index argmax ----------------
  float bestv = -1.0f;  // window sums are >= 0
  int   besti = 0;
#pragma unroll
  for (int j = 0; j < 8; ++j) {
    int i = t * 8 + j;
    int e = i + W - 1;
    if (e > S - 1) e = S - 1;                      // truncated tail windows
    float wv = sq[e] - (i ? sq[i - 1] : 0.f);
    if (wv > bestv) { bestv = wv; besti = i; }     // strict '>' keeps first index
  }
  bv[t] = bestv;
  bi[t] = besti;
  __syncthreads();

  for (int srd = NT / 2; srd > 0; srd >>= 1) {
    if (t < srd) {
      float ov = bv[t + srd];
      int   oi = bi[t + srd];
      if (ov > bv[t] || (ov == bv[t] && oi < bi[t])) { bv[t] = ov; bi[t] = oi; }
    }
    __syncthreads();
  }
  const int best = bi[0];

  // ---------------- Phase 4: gather winning window (64x64 floats) ----------------
  float* ob = out + bf * (size_t)(W * W);
#pragma unroll
  for (int k = 0; k < 4; ++k) {
    int idx4 = t + k * NT;      // 0..1023 float4s
    int w0   = idx4 >> 4;       // output row 0..63
    int w1q  = idx4 & 15;       // float4 within row
    int row  = best + w0;
    if (row > S - 1) row = S - 1;  // clip like take_along_axis
    float4 v = *reinterpret_cast<const float4*>(Xb + (size_t)row * W + w1q * 4);
    *reinterpret_cast<float4*>(ob + idx4 * 4) = v;
  }
}

extern "C" void kernel_launch(void* const* d_in, const int* in_sizes, int n_in,
                              void* d_out, int out_size, void* d_ws, size_t ws_size,
                              hipStream_t stream) {
  (void)in_sizes; (void)n_in; (void)out_size; (void)d_ws; (void)ws_size;
  const float* X = (const float*)d_in[0];
  float* out = (float*)d_out;
  // One workgroup per (b, f) pair: 16 * 128 = 2048 blocks of 256 threads.
  swl2max_kernel<<<dim3(16 * 128), dim3(NT), 0, stream>>>(X, out);
}